// YoloLoss_41343355191628
// MI455X (gfx1250) — compile-verified
//
#include <hip/hip_runtime.h>

#define B_ 64
#define W_ 64
#define H_ 64
#define A_ 5
#define C_ 21
#define M_ 16
#define DS_ 16.0f
#define IOU_T 0.3f

#define BLOCKS_BIG 672
#define THREADS_BIG 256
#define S1_ (BLOCKS_BIG * THREADS_BIG)   // 172032 float4 stride, 4*S1 % 21 == 0
#define S2_ 102400                        // float4 stride for po pass, 4*S2 % 5 == 0
#define NPART BLOCKS_BIG

typedef float v2f __attribute__((ext_vector_type(2)));
typedef float v4f __attribute__((ext_vector_type(4)));
typedef float v8f __attribute__((ext_vector_type(8)));

__device__ __forceinline__ float sigmoidf_(float x) { return 1.0f / (1.0f + __expf(-x)); }

// Workspace float layout:
//  [0..7]   k_obj accumulators: [2]=S_noobj_corr [3]=S_obj [4]=S_prior [5]=S_true [6]=S_score_corr
//  [16 + (b*M+t)*16 ...] : scatter entries {ci, cj, winmask(int), pad, iou[5], box[4]}
//  [16400 .. 16400+671]  : per-block partials of S_score_def
//  [17072 .. 17072+671]  : per-block partials of S_noobj_all
#define ENT_OFF 16
#define P1_OFF 16400
#define P2_OFF 17072

__global__ void k_init(float* ws) {
    if (threadIdx.x < 8) ws[threadIdx.x] = 0.0f;
}

__global__ void k_scatter(const float* __restrict__ po,
                          const float* __restrict__ tb,
                          const float* __restrict__ anch,
                          float* __restrict__ ws) {
    __shared__ int sci[M_], scj[M_], smask[M_];
    const int b = blockIdx.x;
    const int t = threadIdx.x;
    float bx = 0.f, by = 0.f, bw = 0.f, bh = 0.f;
    int ci = 0, cj = 0, mask = 0;
    float iou[A_] = {0.f, 0.f, 0.f, 0.f, 0.f};
    if (t < M_) {
        const float* p = tb + (b * M_ + t) * 4;
        bx = p[0] * (1.0f / DS_);
        by = p[1] * (1.0f / DS_);
        bw = p[2] * (1.0f / DS_);
        bh = p[3] * (1.0f / DS_);
        ci = (int)bx;
        cj = (int)by;
        const int cell = (b * W_ + ci) * H_ + cj;
#pragma unroll
        for (int a = 0; a < A_; ++a) {
            const float* q = po + ((long long)(cell * A_ + a)) * 5;
            float pw = __expf(q[2]) * anch[2 * a + 0];
            float ph = __expf(q[3]) * anch[2 * a + 1];
            float inter = fminf(pw, bw) * fminf(ph, bh);
            float uni = pw * ph + bw * bh - inter;
            float v = inter / uni;
            iou[a] = v;
            if (v > IOU_T) mask |= (1 << a);
        }
        sci[t] = ci; scj[t] = cj; smask[t] = mask;
    }
    __syncthreads();
    if (t < M_) {
        // last-writer-wins per (cell, anchor)
        int win = mask;
        for (int t2 = t + 1; t2 < M_; ++t2)
            if (sci[t2] == ci && scj[t2] == cj) win &= ~smask[t2];
        float* e = ws + ENT_OFF + (b * M_ + t) * 16;
        e[0] = (float)ci;
        e[1] = (float)cj;
        ((int*)e)[2] = win;
        e[3] = 0.f;
#pragma unroll
        for (int a = 0; a < A_; ++a) e[4 + a] = iou[a];
        e[9] = bx; e[10] = by; e[11] = bw; e[12] = bh;
    }
}

// Exactly B_*M_ = 1024 threads (4 blocks x 256) -> no bounds checks, no early
// returns; every lane reaches the shuffle reduction.
__global__ void k_obj(const float* __restrict__ cls,
                      const float* __restrict__ po,
                      const float* __restrict__ labels,
                      const float* __restrict__ anch,
                      float* __restrict__ ws) {
    const int idx = blockIdx.x * blockDim.x + threadIdx.x;  // 0..1023
    const int b = idx >> 4, t = idx & (M_ - 1);
    const float* e = ws + ENT_OFF + idx * 16;
    const int win = ((const int*)e)[2];
    float lno = 0.f, lobj = 0.f, lpr = 0.f, ltr = 0.f, lsc = 0.f;
    if (win) {
        const int ci = (int)e[0], cj = (int)e[1];
        const float bx = e[9], by = e[10], bw = e[11], bh = e[12];
        const int cell = (b * W_ + ci) * H_ + cj;
        for (int a = 0; a < A_; ++a) {
            if (!((win >> a) & 1)) continue;
            const float* q = po + ((long long)(cell * A_ + a)) * 5;
            float px = sigmoidf_(q[0]) + (float)ci;
            float py = sigmoidf_(q[1]) + (float)cj;
            float ax = anch[2 * a], ay = anch[2 * a + 1];
            float pw = __expf(q[2]) * ax;
            float ph = __expf(q[3]) * ay;
            float conf = sigmoidf_(q[4]);
            float iou = e[4 + a];
            float dob = conf - iou;
            lobj += 0.5f * dob * dob;
            lno += 0.5f * conf * conf;
            float dw = pw - ax, dh = ph - ay;
            lpr += 0.5f * (dw * dw + dh * dh);
            float d0 = px - bx, d1 = py - by, d2 = pw - bw, d3 = ph - bh;
            ltr += 0.5f * (d0 * d0 + d1 * d1 + d2 * d2 + d3 * d3);
            const float* cp = cls + ((long long)(cell * A_ + a)) * C_;
            const float* lp = labels + (b * M_ + t) * C_;
            for (int c = 0; c < C_; ++c) {
                float x = cp[c];
                float l = lp[c];
                float d = (c == C_ - 1) ? 1.0f : 0.0f;
                float e1 = x - l, e2 = x - d;
                lsc += 0.5f * (e1 * e1 - e2 * e2);
            }
        }
    }
    // wave reduction -> 5 atomics per wave instead of per thread
#pragma unroll
    for (int o = 16; o > 0; o >>= 1) {
        lno  += __shfl_down(lno, o);
        lobj += __shfl_down(lobj, o);
        lpr  += __shfl_down(lpr, o);
        ltr  += __shfl_down(ltr, o);
        lsc  += __shfl_down(lsc, o);
    }
    if ((threadIdx.x & 31) == 0) {
        atomicAdd(ws + 2, lno);
        atomicAdd(ws + 3, lobj);
        atomicAdd(ws + 4, lpr);
        atomicAdd(ws + 5, ltr);
        atomicAdd(ws + 6, lsc);
    }
}

// Streaming pass over cls_score (110 MB) and pred_object (26 MB).
// Grid chosen so element counts divide exactly: no predication anywhere,
// EXEC is all-ones for every in-loop V_WMMA_F32_16X16X4_F32, and the %21 / %5
// channel residues are per-thread loop invariants. 8 independent b128 loads
// per unrolled body keep ~22 MB in flight across 5376 waves.
__global__ void __launch_bounds__(THREADS_BIG)
k_big(const float* __restrict__ cls,
      const float* __restrict__ po,
      float* __restrict__ ws) {
    const int tid = blockIdx.x * THREADS_BIG + threadIdx.x;
    const int lane = threadIdx.x & 31;
    const int wave = threadIdx.x >> 5;

    v2f ones; ones.x = 1.0f; ones.y = 1.0f;
    v8f accA = {}, accB = {}, acc2 = {};

    // ---- phase 1: 0.5*(cls - default)^2, N4 = 6,881,280 float4 = 40 * S1_
    {
        const v4f* cls4 = (const v4f*)cls;
        const unsigned r = (4u * (unsigned)tid) % 21u;  // loop-invariant residue
        const float d0 = (r == 20u) ? 1.0f : 0.0f;
        const float d1 = (r == 19u) ? 1.0f : 0.0f;
        const float d2 = (r == 18u) ? 1.0f : 0.0f;
        const float d3 = (r == 17u) ? 1.0f : 0.0f;
        int idx = tid;
        for (int k = 0; k < 5; ++k) {
            v4f v[8];
#pragma unroll
            for (int u = 0; u < 8; ++u) v[u] = cls4[idx + u * S1_];
#pragma unroll
            for (int u = 0; u < 8; ++u) {
                float x0 = v[u].x - d0, x1 = v[u].y - d1;
                float x2 = v[u].z - d2, x3 = v[u].w - d3;
                v2f a;
                a.x = __builtin_fmaf(x0, x0, x1 * x1);
                a.y = __builtin_fmaf(x2, x2, x3 * x3);
                if (u & 1)
                    accB = __builtin_amdgcn_wmma_f32_16x16x4_f32(false, a, false, ones,
                                                                 (short)0, accB, false, false);
                else
                    accA = __builtin_amdgcn_wmma_f32_16x16x4_f32(false, a, false, ones,
                                                                 (short)0, accA, false, false);
            }
            idx += 8 * S1_;
        }
    }

    // ---- phase 2: 0.5*conf^2 over pred_object[...,4]; Npo4 = 1,638,400 = 16 * S2_
    float p2 = 0.0f;
    if (tid < S2_) {  // first 400 whole blocks; no intra-wave divergence
        const v4f* po4 = (const v4f*)po;
        const unsigned r5 = (4u * (unsigned)tid) % 5u;  // loop-invariant
        const float have = (r5 != 0u) ? 1.0f : 0.0f;
        int idx = tid;
        for (int k = 0; k < 2; ++k) {
            v4f w[8];
#pragma unroll
            for (int u = 0; u < 8; ++u) w[u] = po4[idx + u * S2_];
#pragma unroll
            for (int u = 0; u < 8; ++u) {
                float z = (r5 == 1u) ? w[u].w
                        : (r5 == 2u) ? w[u].z
                        : (r5 == 3u) ? w[u].y
                                     : w[u].x;  // r5==4 -> .x ; r5==0 masked
                float c = sigmoidf_(z);
                p2 = __builtin_fmaf(have * c, c, p2);
            }
            idx += 8 * S2_;
        }
    }
    {   // fold p2 through the XDL reduction as well (EXEC all-ones here)
        v2f a2; a2.x = p2; a2.y = 0.0f;
        acc2 = __builtin_amdgcn_wmma_f32_16x16x4_f32(false, a2, false, ones,
                                                     (short)0, acc2, false, false);
    }

    float s1 = 0.f, s2 = 0.f;
#pragma unroll
    for (int k = 0; k < 8; ++k) { s1 += accA[k] + accB[k]; s2 += acc2[k]; }

    // wave totals live in column n=0: lanes 0 (M rows 0..7) and 16 (rows 8..15)
    __shared__ float l1[16], l2[16];
    if (lane == 0)  { l1[wave * 2 + 0] = s1; l2[wave * 2 + 0] = s2; }
    if (lane == 16) { l1[wave * 2 + 1] = s1; l2[wave * 2 + 1] = s2; }
    __syncthreads();
    if (threadIdx.x == 0) {
        float t1 = 0.f, t2 = 0.f;
#pragma unroll
        for (int i = 0; i < 16; ++i) { t1 += l1[i]; t2 += l2[i]; }
        ws[P1_OFF + blockIdx.x] = 0.5f * t1;   // no atomics: one slot per block
        ws[P2_OFF + blockIdx.x] = 0.5f * t2;
    }
}

__global__ void k_final(const float* __restrict__ ws,
                        const int* __restrict__ iter,
                        float* __restrict__ out) {
    const int lane = threadIdx.x;  // single wave of 32
    float t1 = 0.f, t2 = 0.f;
    for (int i = lane; i < NPART; i += 32) {   // 672 = 32 * 21
        t1 += ws[P1_OFF + i];
        t2 += ws[P2_OFF + i];
    }
#pragma unroll
    for (int o = 16; o > 0; o >>= 1) {
        t1 += __shfl_down(t1, o);
        t2 += __shfl_down(t2, o);
    }
    if (lane == 0) {
        const float Nb = (float)(B_ * W_ * H_ * A_);
        const float need = (iter[0] < 1000) ? 1.0f : 0.0f;
        float noobj = (t2 - ws[2]) / Nb;                 // SCALE_NOOBJ = 1
        float obj   = 5.0f * ws[3] / Nb;                 // SCALE_OBJ  = 5
        float prior = need * 5.0f * ws[4] / (Nb * 2.0f);
        float truel = 5.0f * ws[5] / (Nb * 4.0f);
        float score = 5.0f * (t1 + ws[6]) / (Nb * (float)C_);
        out[0] = 0.25f * (noobj + obj + prior + truel + score);
    }
}

extern "C" void kernel_launch(void* const* d_in, const int* in_sizes, int n_in,
                              void* d_out, int out_size, void* d_ws, size_t ws_size,
                              hipStream_t stream) {
    const float* cls    = (const float*)d_in[0];  // (B,W,H,A,C)
    const float* po     = (const float*)d_in[1];  // (B,W,H,A,5)
    const float* tb     = (const float*)d_in[2];  // (B,M,4)
    const float* labels = (const float*)d_in[3];  // (B,M,C)
    const float* anch   = (const float*)d_in[4];  // (A,2)
    const int*   iter   = (const int*)d_in[5];    // scalar
    float* ws  = (float*)d_ws;                    // ~71 KB used
    float* out = (float*)d_out;

    k_init<<<1, 32, 0, stream>>>(ws);
    k_scatter<<<B_, 32, 0, stream>>>(po, tb, anch, ws);
    k_obj<<<(B_ * M_) / 256, 256, 0, stream>>>(cls, po, labels, anch, ws);
    k_big<<<BLOCKS_BIG, THREADS_BIG, 0, stream>>>(cls, po, ws);
    k_final<<<1, 32, 0, stream>>>(ws, iter, out);
}